// Quantizer_60206851555633
// MI455X (gfx1250) — compile-verified
//
#include <hip/hip_runtime.h>

#define CODEBOOK 512
#define QBLOCK 256
#define LDS_PADDED 576   // pad_idx(511)+1 = 572, rounded up

typedef __attribute__((ext_vector_type(4))) float float4v;
typedef __attribute__((ext_vector_type(4))) int   int4v;

#if defined(__has_builtin)
# if __has_builtin(__builtin_amdgcn_global_load_async_to_lds_b128)
#  define USE_ASYNC_LDS 1
# endif
#endif

#ifdef USE_ASYNC_LDS
typedef __attribute__((address_space(1))) int4v* g4p_t;  // global int4*
typedef __attribute__((address_space(3))) int4v* l4p_t;  // LDS int4*
#endif

// Insert a 4-float pad every 32 floats: keeps 16B alignment for b128 staging,
// spreads the binary-search probe sets (idx ≡ -1 mod step) across LDS banks.
__device__ __forceinline__ unsigned pad_idx(unsigned i) {
  return i + ((i >> 5) << 2);
}

// Monotone float -> uint mapping: a < b (IEEE, no NaN) <=> map(a) < map(b).
__device__ __forceinline__ unsigned fmap(float f) {
  unsigned u = __float_as_uint(f);
  return u ^ ((u & 0x80000000u) ? 0xFFFFFFFFu : 0x80000000u);
}

// ---------------------------------------------------------------------------
// Kernel 1: O(K^2) rank sort of the 512-entry codebook into d_ws.
// One 512-thread block, no barriers. key_j is uniform -> built in SALU from
// scalar loads; the per-lane cost is 1x v_cmp_lt_u64 + 1 add per iteration.
// ---------------------------------------------------------------------------
__global__ __launch_bounds__(CODEBOOK) void sort_codes_kernel(
    const float* __restrict__ emb, float* __restrict__ sorted) {
  const int t   = threadIdx.x;        // 0..511
  const float c = emb[t];
  const unsigned long long keyt =
      (((unsigned long long)fmap(c)) << 16) | (unsigned)t;
  int rank = 0;
#pragma unroll 8
  for (int j = 0; j < CODEBOOK; ++j) {
    const unsigned long long keyj =
        (((unsigned long long)fmap(emb[j])) << 16) | (unsigned)j;  // SALU
    rank += (keyj < keyt);
  }
  sorted[rank] = c;
}

// ---------------------------------------------------------------------------
// Kernel 2: nearest-code lookup via binary search over sorted codes in LDS.
// One float4 per lane; codebook staged with gfx1250 async global->LDS copy
// into a bank-conflict-padded layout. Top 3 tree levels held in registers.
// ---------------------------------------------------------------------------
__global__ __launch_bounds__(QBLOCK) void quantize_kernel(
    const float* __restrict__ h, const float* __restrict__ sorted,
    float* __restrict__ out, int n) {
  __shared__ float codes[LDS_PADDED];
  const unsigned t   = threadIdx.x;
  const unsigned gid = blockIdx.x * QBLOCK + t;   // float4 index

  // Stage the sorted codebook into padded LDS (async global->LDS, ASYNCcnt).
#ifdef USE_ASYNC_LDS
  if (t < (CODEBOOK / 4)) {
    float* gsrc = const_cast<float*>(sorted) + t * 4;
    float* ldst = codes + pad_idx(t * 4);
    __builtin_amdgcn_global_load_async_to_lds_b128(
        (g4p_t)gsrc, (l4p_t)ldst, /*imm offset*/ 0, /*cpol*/ 0);
  }
#else
  if (t < (CODEBOOK / 4)) {
    *(float4v*)(codes + pad_idx(t * 4)) = ((const float4v*)sorted)[t];
  }
#endif

  // Issue the streaming h load early so it overlaps codebook staging.
  const bool active = (int)(gid * 4) < n;
  float4v x4 = {0.f, 0.f, 0.f, 0.f};
  if (active) x4 = __builtin_nontemporal_load(&((const float4v*)h)[gid]);

#ifdef USE_ASYNC_LDS
# if __has_builtin(__builtin_amdgcn_s_wait_asynccnt)
  __builtin_amdgcn_s_wait_asynccnt(0);
# else
  asm volatile("s_wait_asynccnt 0" ::: "memory");
# endif
#endif
  __syncthreads();

  // Top 3 levels of the search tree: 7 values, same address across all lanes
  // (LDS broadcast reads, conflict-free), amortized over 4 elements.
  const float c255 = codes[pad_idx(255)];
  const float c127 = codes[pad_idx(127)];
  const float c383 = codes[pad_idx(383)];
  const float c63  = codes[pad_idx(63)];
  const float c191 = codes[pad_idx(191)];
  const float c319 = codes[pad_idx(319)];
  const float c447 = codes[pad_idx(447)];

  float r[4];
#pragma unroll
  for (int e = 0; e < 4; ++e) {
    const float x = x4[e];
    // Levels step=256,128,64 resolved from registers (cndmask selects).
    const bool b2 = (c255 < x);
    unsigned pos = b2 ? 256u : 0u;
    const float cA = b2 ? c383 : c127;
    const bool b1 = (cA < x);
    pos += b1 ? 128u : 0u;
    const float t1 = b2 ? c447 : c191;
    const float t0 = b2 ? c319 : c63;
    const float cB = b1 ? t1 : t0;
    pos += (cB < x) ? 64u : 0u;
    // Levels step=32..1: 6 LDS probes (padded layout -> spread banks).
#pragma unroll
    for (unsigned step = 32; step > 0; step >>= 1) {
      const unsigned next = pos + step;
      pos = (codes[pad_idx(next - 1)] < x) ? next : pos;
    }
    // pos = count of codes < x. Nearest is codes[pos-1] or codes[pos] (clamped).
    const float cl = codes[pad_idx((pos == 0) ? 0u : (pos - 1u))];
    const float ch = codes[pad_idx((pos == CODEBOOK) ? (CODEBOOK - 1u) : pos)];
    r[e] = ((x - cl) <= (ch - x)) ? cl : ch;
  }

  if (active) {
    float4v o = {r[0], r[1], r[2], r[3]};
    __builtin_nontemporal_store(o, &((float4v*)out)[gid]);
  }
}

// ---------------------------------------------------------------------------
extern "C" void kernel_launch(void* const* d_in, const int* in_sizes, int n_in,
                              void* d_out, int out_size, void* d_ws, size_t ws_size,
                              hipStream_t stream) {
  const float* h   = (const float*)d_in[0];   // (8,128,32,32) f32, flat n
  const float* emb = (const float*)d_in[1];   // (512,1) f32
  float* out       = (float*)d_out;
  float* sorted    = (float*)d_ws;            // 512 floats scratch

  const int n = in_sizes[0];                  // 1,048,576

  sort_codes_kernel<<<1, CODEBOOK, 0, stream>>>(emb, sorted);

  const int vec4  = (n + 3) / 4;
  const int grid  = (vec4 + QBLOCK - 1) / QBLOCK;   // 1024 blocks for this size
  quantize_kernel<<<grid, QBLOCK, 0, stream>>>(h, sorted, out, n);
}